// DMPNN_3934190044027
// MI455X (gfx1250) — compile-verified
//
#include <hip/hip_runtime.h>
#include <math.h>

#define NN   50000
#define NE   400000
#define NLE  1600000
#define NB   256
#define NF   64
#define NED  16
#define NITER 3
#define NSF  384

#define CDIV(a,b) (((a)+(b)-1)/(b))
#define ORD_NEG_INF 0x007FFFFFu

typedef float v2f __attribute__((ext_vector_type(2)));
typedef float v8f __attribute__((ext_vector_type(8)));

__device__ __forceinline__ unsigned f2ord(float f) {
  unsigned u = __float_as_uint(f);
  return (u & 0x80000000u) ? ~u : (u | 0x80000000u);
}
__device__ __forceinline__ float ord2f(unsigned u) {
  return (u & 0x80000000u) ? __uint_as_float(u & 0x7fffffffu) : __uint_as_float(~u);
}

// ---------------------------------------------------------------------------
// f32 WMMA GEMM: C[M,Nc] = A[M,K] @ B[K,Nc] (+ bias[Nc]).
// One wave -> 16x64 output strip (4 accumulators, A frag reused 4x).
// Requires M%16==0, K%4==0, Nc%64==0 (all shapes here satisfy this).
// ---------------------------------------------------------------------------
__global__ __launch_bounds__(256)
void gemm_wmma_f32(const float* __restrict__ A, const float* __restrict__ B,
                   const float* __restrict__ bias, float* __restrict__ C,
                   int M, int K, int Nc) {
  int gw   = (int)((blockIdx.x * blockDim.x + threadIdx.x) >> 5);
  int lane = threadIdx.x & 31;
  int tn = Nc >> 6, tm = M >> 4;
  if (gw >= tm * tn) return;             // whole-wave exit: EXEC stays all-ones
  int mt = gw / tn, nt = gw % tn;
  int row0 = mt << 4, col0 = nt << 6;
  int half = lane >> 4, lr = lane & 15;

  v8f ac0 = {}, ac1 = {}, ac2 = {}, ac3 = {};
  const float* Ar = A + (size_t)(row0 + lr) * K + (half << 1);
  for (int k = 0; k < K; k += 4) {
    v2f a; a.x = Ar[k]; a.y = Ar[k + 1];
    const float* Bp = B + (size_t)(k + (half << 1)) * Nc + col0 + lr;
    v2f b0, b1, b2, b3;
    b0.x = Bp[0];  b0.y = Bp[Nc];
    b1.x = Bp[16]; b1.y = Bp[Nc + 16];
    b2.x = Bp[32]; b2.y = Bp[Nc + 32];
    b3.x = Bp[48]; b3.y = Bp[Nc + 48];
    ac0 = __builtin_amdgcn_wmma_f32_16x16x4_f32(false, a, false, b0, (short)0, ac0, false, false);
    ac1 = __builtin_amdgcn_wmma_f32_16x16x4_f32(false, a, false, b1, (short)0, ac1, false, false);
    ac2 = __builtin_amdgcn_wmma_f32_16x16x4_f32(false, a, false, b2, (short)0, ac2, false, false);
    ac3 = __builtin_amdgcn_wmma_f32_16x16x4_f32(false, a, false, b3, (short)0, ac3, false, false);
  }
  float bb0 = 0.f, bb1 = 0.f, bb2 = 0.f, bb3 = 0.f;
  if (bias) {
    bb0 = bias[col0 + lr];      bb1 = bias[col0 + lr + 16];
    bb2 = bias[col0 + lr + 32]; bb3 = bias[col0 + lr + 48];
  }
  int rb = row0 + (half << 3);
  for (int j = 0; j < 8; ++j) {
    size_t o = (size_t)(rb + j) * Nc + col0 + lr;
    C[o]      = ac0[j] + bb0;
    C[o + 16] = ac1[j] + bb1;
    C[o + 32] = ac2[j] + bb2;
    C[o + 48] = ac3[j] + bb3;
  }
}

static inline void gemm(const float* A, const float* B, const float* bias, float* C,
                        int M, int K, int Nc, hipStream_t s) {
  int waves = (M >> 4) * (Nc >> 6);
  gemm_wmma_f32<<<CDIV(waves, 8), 256, 0, s>>>(A, B, bias, C, M, K, Nc);
}

// ---------------------------------------------------------------------------
// CSR build for the line graph (keyed by destination edge lg1)
// ---------------------------------------------------------------------------
__global__ void lg_hist(const int* __restrict__ lg1, int* __restrict__ deg) {
  int i = blockIdx.x * 256 + threadIdx.x;
  if (i < NLE) atomicAdd(&deg[lg1[i]], 1);
}

// single-workgroup exclusive scan: rowptr[0..NE], rowptr[NE] = NLE
__global__ __launch_bounds__(1024)
void lg_scan(const int* __restrict__ deg, int* __restrict__ rowptr) {
  __shared__ int buf[1024];
  __shared__ int carry;
  int tid = threadIdx.x;
  if (tid == 0) carry = 0;
  __syncthreads();
  for (int base = 0; base < NE; base += 1024) {
    int v = (base + tid < NE) ? deg[base + tid] : 0;
    buf[tid] = v;
    __syncthreads();
    for (int off = 1; off < 1024; off <<= 1) {
      int t = (tid >= off) ? buf[tid - off] : 0;
      __syncthreads();
      buf[tid] += t;
      __syncthreads();
    }
    if (base + tid < NE) rowptr[base + tid] = carry + buf[tid] - v;
    __syncthreads();
    if (tid == 0) carry += buf[1023];
    __syncthreads();
  }
  if (tid == 0) rowptr[NE] = carry;
}

// cursor starts as a copy of rowptr; csr_src[pos] = source edge of each line edge
__global__ void lg_fill(const int* __restrict__ lg0, const int* __restrict__ lg1,
                        int* __restrict__ cursor, int* __restrict__ csr_src) {
  int i = blockIdx.x * 256 + threadIdx.x;
  if (i >= NLE) return;
  int pos = atomicAdd(&cursor[lg1[i]], 1);
  csr_src[pos] = lg0[i];
}

// out_t[e,f] = ea[e,f] + sum_{s in CSR row e} prev[s,f]   (pure gather, no atomics)
__global__ void gather_lg(const float* __restrict__ prev, const float* __restrict__ ea,
                          float* __restrict__ out_t, const int* __restrict__ rowptr,
                          const int* __restrict__ csr_src) {
  int idx = blockIdx.x * 256 + threadIdx.x;
  if (idx >= NE * NF) return;
  int e = idx >> 6, f = idx & 63;
  int beg = rowptr[e], end = rowptr[e + 1];
  float acc = ea[idx];
  for (int j = beg; j < end; ++j) {
    int s = csr_src[j];
    if (j + 1 < end)  // prefetch next gathered row (global_prefetch_b8)
      __builtin_prefetch(prev + (size_t)csr_src[j + 1] * NF + f, 0, 3);
    acc += prev[(size_t)s * NF + f];
  }
  out_t[idx] = acc;
}

// ---------------------------------------------------------------------------
// Graph / elementwise kernels
// ---------------------------------------------------------------------------
__global__ void fill_u32(unsigned* p, unsigned v, int n) {
  int i = blockIdx.x * 256 + threadIdx.x;
  if (i < n) p[i] = v;
}

// ea[e,f] = (XWu[ei0[e],f] + XWv[ei1[e],f] + ea[e,f]) / 3   (ea pre-holds edge_attr@W_e)
__global__ void build_ea(const float* __restrict__ XWu, const float* __restrict__ XWv,
                         const int* __restrict__ ei0, const int* __restrict__ ei1,
                         float* __restrict__ ea) {
  int idx = blockIdx.x * 256 + threadIdx.x;
  if (idx >= NE * NF) return;
  int e = idx >> 6, f = idx & 63;
  ea[idx] = (XWu[(size_t)ei0[e] * NF + f] + XWv[(size_t)ei1[e] * NF + f] + ea[idx]) * (1.f / 3.f);
}

// xw[e] = out[e,:] . gat_W   (one wave per edge, coalesced, shfl reduce)
__global__ void edge_dot(const float* __restrict__ X, const float* __restrict__ w64,
                         float* __restrict__ dotv) {
  int wid = (int)((blockIdx.x * blockDim.x + threadIdx.x) >> 5);
  int lane = threadIdx.x & 31;
  if (wid >= NE) return;
  const float* row = X + (size_t)wid * NF;
  float v = row[lane] * w64[lane] + row[lane + 32] * w64[lane + 32];
  for (int off = 16; off; off >>= 1) v += __shfl_xor(v, off, 32);
  if (lane == 0) dotv[wid] = v;
}

// pass 1: segment max of leaky-relu attention logits over dst (line edges + self loops)
__global__ void gat_max(const float* __restrict__ xw, const float* as_p, const float* ad_p,
                        const int* __restrict__ lg0, const int* __restrict__ lg1,
                        unsigned* __restrict__ m_u) {
  int i = blockIdx.x * 256 + threadIdx.x;
  if (i >= NLE + NE) return;
  float as = as_p[0], ad = ad_p[0];
  int s, d;
  if (i < NLE) { s = lg0[i]; d = lg1[i]; } else { s = d = i - NLE; }
  float al = xw[s] * as + xw[d] * ad;
  al = al >= 0.f ? al : 0.2f * al;
  atomicMax(&m_u[d], f2ord(al));
}

// pass 2: s[d] += exp(al-m[d]);  num[d] += exp(al-m[d]) * xw[s]
__global__ void gat_sum(const float* __restrict__ xw, const float* as_p, const float* ad_p,
                        const int* __restrict__ lg0, const int* __restrict__ lg1,
                        const unsigned* __restrict__ m_u, float* __restrict__ ssum,
                        float* __restrict__ num) {
  int i = blockIdx.x * 256 + threadIdx.x;
  if (i >= NLE + NE) return;
  float as = as_p[0], ad = ad_p[0];
  int s, d;
  if (i < NLE) { s = lg0[i]; d = lg1[i]; } else { s = d = i - NLE; }
  float al = xw[s] * as + xw[d] * ad;
  al = al >= 0.f ? al : 0.2f * al;
  float e = expf(al - ord2f(m_u[d]));
  atomicAdd(&ssum[d], e);
  atomicAdd(&num[d], e * xw[s]);
}

// xc[e] = num/(s+eps) + bias ; pool max per batch
__global__ void gat_xc_poolmax(const float* __restrict__ num, const float* __restrict__ ssum,
                               const float* gbias, const int* __restrict__ batch,
                               float* __restrict__ xcb, unsigned* __restrict__ mb) {
  int e = blockIdx.x * 256 + threadIdx.x;
  if (e >= NE) return;
  float xc = num[e] / (ssum[e] + 1e-16f) + gbias[0];
  xcb[e] = xc;
  atomicMax(&mb[batch[e]], f2ord(xc));
}

__global__ void gat_poolw(const float* __restrict__ xcb, const unsigned* __restrict__ mb,
                          const int* __restrict__ batch, float* __restrict__ wexp,
                          float* __restrict__ sb) {
  int e = blockIdx.x * 256 + threadIdx.x;
  if (e >= NE) return;
  int b = batch[e];
  float wv = expf(xcb[e] - ord2f(mb[b]));
  wexp[e] = wv;
  atomicAdd(&sb[b], wv);
}

// gout_raw[b,f] += wexp[e]*out[e,f]; batch is sorted -> register accumulate, flush on change
__global__ void gat_accum(const float* __restrict__ out_t, const float* __restrict__ wexp,
                          const int* __restrict__ batch, float* __restrict__ gout_raw) {
  int f = threadIdx.x & 63, slot = threadIdx.x >> 6;
  int e0 = blockIdx.x * 128;
  float acc = 0.f; int cur = -1;
  for (int e = e0 + slot; e < e0 + 128 && e < NE; e += 4) {
    int b = batch[e];
    if (b != cur) {
      if (cur >= 0) atomicAdd(&gout_raw[(size_t)cur * NF + f], acc);
      acc = 0.f; cur = b;
    }
    acc += wexp[e] * out_t[(size_t)e * NF + f];
  }
  if (cur >= 0) atomicAdd(&gout_raw[(size_t)cur * NF + f], acc);
}

__global__ void gat_fin(float* __restrict__ gout_raw, const float* __restrict__ sb) {
  int idx = blockIdx.x * 256 + threadIdx.x;
  if (idx >= NB * NF) return;
  gout_raw[idx] /= (sb[idx >> 6] + 1e-16f);
}

__global__ void tanh_k(float* __restrict__ X, int n) {
  int i = blockIdx.x * 256 + threadIdx.x;
  if (i < n) X[i] = tanhf(X[i]);
}

// per-graph softmax over the 3 iterations: sc[b,t]
__global__ void attn_sc(const float* __restrict__ gouts, const float* __restrict__ aw,
                        const float* __restrict__ ab, float* __restrict__ scb) {
  int b = blockIdx.x * 256 + threadIdx.x;
  if (b >= NB) return;
  float lg[NITER];
  float mx = -1e30f;
  for (int t = 0; t < NITER; ++t) {
    float s = 0.f;
    for (int f = 0; f < NF; ++f)
      s += gouts[(size_t)t * NB * NF + (size_t)b * NF + f] * aw[f * NITER + t];
    s += ab[t];
    lg[t] = s;
    mx = fmaxf(mx, s);
  }
  float den = 0.f;
  for (int t = 0; t < NITER; ++t) { lg[t] = expf(lg[t] - mx); den += lg[t]; }
  for (int t = 0; t < NITER; ++t) scb[b * NITER + t] = lg[t] / den;
}

// xn[ei1[e],f] += sum_t outs[t][e,f] * sc[batch[e],t]
__global__ void final_edge_scatter(const float* __restrict__ outs, const float* __restrict__ scb,
                                   const int* __restrict__ batch, const int* __restrict__ ei1,
                                   float* __restrict__ xn) {
  int idx = blockIdx.x * 256 + threadIdx.x;
  if (idx >= NE * NF) return;
  int e = idx >> 6, f = idx & 63;
  int b = batch[e];
  float v = outs[idx]                       * scb[b * NITER + 0]
          + outs[(size_t)NE * NF + idx]     * scb[b * NITER + 1]
          + outs[(size_t)2 * NE * NF + idx] * scb[b * NITER + 2];
  atomicAdd(&xn[(size_t)ei1[e] * NF + f], v);
}

// BatchNorm column moments: 64 columns per block-x, 128-row stripes per block-y
__global__ void bn_stats(const float* __restrict__ X, int Nr, int C,
                         float* __restrict__ s1, float* __restrict__ s2) {
  int f = threadIdx.x & 63, slot = threadIdx.x >> 6;
  int c = blockIdx.x * 64 + f;
  int r0 = blockIdx.y * 128;
  int rend = r0 + 128 < Nr ? r0 + 128 : Nr;
  float a = 0.f, b = 0.f;
  for (int r = r0 + slot; r < rend; r += 4) {
    float v = X[(size_t)r * C + c];
    a += v; b += v * v;
  }
  atomicAdd(&s1[c], a);
  atomicAdd(&s2[c], b);
}

__global__ void bn_finalize(const float* __restrict__ s1, const float* __restrict__ s2,
                            const float* __restrict__ g, const float* __restrict__ bta,
                            int Nr, int C, float* __restrict__ scale, float* __restrict__ shift) {
  int c = blockIdx.x * 256 + threadIdx.x;
  if (c >= C) return;
  float m   = s1[c] / (float)Nr;
  float var = s2[c] / (float)Nr - m * m;
  float inv = rsqrtf(var + 1e-5f);
  float sc = g[c] * inv;
  scale[c] = sc;
  shift[c] = bta[c] - m * sc;
}

// fold BN affine into lin1: W1f[k,n] = scale[k]*W1[k,n]; b1f[n] = b1[n] + sum_k shift[k]*W1[k,n]
__global__ void fold_w1(const float* __restrict__ scale, const float* __restrict__ shift,
                        const float* __restrict__ W1, const float* __restrict__ b1,
                        float* __restrict__ W1f, float* __restrict__ b1f) {
  int n = blockIdx.x * 256 + threadIdx.x;
  if (n >= NSF) return;
  float acc = b1[n];
  for (int k = 0; k < NF; ++k) {
    float w = W1[(size_t)k * NSF + n];
    W1f[(size_t)k * NSF + n] = scale[k] * w;
    acc += shift[k] * w;
  }
  b1f[n] = acc;
}

__global__ void bn_prelu(const float* __restrict__ X, const float* __restrict__ scale,
                         const float* __restrict__ shift, const float* __restrict__ pw, int pidx,
                         float* __restrict__ Y, int total, int C) {
  int idx = blockIdx.x * 256 + threadIdx.x;
  if (idx >= total) return;
  int c = idx % C;
  float v = X[idx] * scale[c] + shift[c];
  float p = pw[pidx];
  Y[idx] = v >= 0.f ? v : p * v;
}

__global__ void resid_avg(float* __restrict__ Xa, const float* __restrict__ Xb, int total) {
  int idx = blockIdx.x * 256 + threadIdx.x;
  if (idx >= total) return;
  Xa[idx] = (Xa[idx] + Xb[idx]) * 0.5f;
}

// ---------------------------------------------------------------------------
extern "C" void kernel_launch(void* const* d_in, const int* in_sizes, int n_in,
                              void* d_out, int out_size, void* d_ws, size_t ws_size,
                              hipStream_t stream) {
  const float* x         = (const float*)d_in[0];
  const float* edge_attr = (const float*)d_in[1];
  const int*   ei        = (const int*)d_in[2];
  const int*   lg        = (const int*)d_in[3];
  const int*   batch     = (const int*)d_in[4];
  const float* W_u = (const float*)d_in[5];
  const float* W_v = (const float*)d_in[6];
  const float* W_e = (const float*)d_in[7];
  const float* gat_W   = (const float*)d_in[8];
  const float* att_src = (const float*)d_in[9];
  const float* att_dst = (const float*)d_in[10];
  const float* gat_bias= (const float*)d_in[11];
  const float* a_w    = (const float*)d_in[12];
  const float* a_bias = (const float*)d_in[13];
  const float* W_gout = (const float*)d_in[14];
  const float* b_gout = (const float*)d_in[15];
  const float* bn1_g = (const float*)d_in[16];
  const float* bn1_b = (const float*)d_in[17];
  const float* W1 = (const float*)d_in[18];
  const float* b1 = (const float*)d_in[19];
  const float* bn_g = (const float*)d_in[20];
  const float* bn_b = (const float*)d_in[21];
  const float* pr  = (const float*)d_in[22];
  const float* Wm  = (const float*)d_in[23];
  const float* bm  = (const float*)d_in[24];
  const float* bn5_g = (const float*)d_in[25];
  const float* bn5_b = (const float*)d_in[26];
  const float* pr5 = (const float*)d_in[27];
  const float* W5  = (const float*)d_in[28];
  const float* b5  = (const float*)d_in[29];
  const int* ei0 = ei,  *ei1 = ei + NE;
  const int* lg0 = lg,  *lg1 = lg + NLE;

  // workspace carve-up (floats)
  float* w = (float*)d_ws;
  float* XWu = w;  w += (size_t)NN * NF;
  float* XWv = w;  w += (size_t)NN * NF;
  float* ea  = w;  w += (size_t)NE * NF;
  float* outs = w; w += (size_t)NITER * NE * NF;
  float*    xw   = w;            w += NE;
  unsigned* m_u  = (unsigned*)w; w += NE;
  float*    ssum = w;            w += NE;
  float*    num  = w;            w += NE;
  float*    xcb  = w;            w += NE;
  float*    wexp = w;            w += NE;
  unsigned* mb = (unsigned*)w;   w += NB;
  float* sb = w;                 w += NB;
  float* gout_raw = w;           w += (size_t)NB * NF;
  float* gouts = w;              w += (size_t)NITER * NB * NF;
  float* scb = w;                w += (size_t)NB * NITER;
  float* xn = w;                 w += (size_t)NN * NF;
  float* X1 = w;                 w += (size_t)NN * NSF;
  float* ACT = w;                w += (size_t)NN * NSF;
  float* Hb = w;                 w += (size_t)NN * NSF;
  float* W1f = w;                w += (size_t)NF * NSF;
  float* b1f = w;                w += NSF;
  float* cs1 = w;                w += NSF;
  float* cs2 = w;                w += NSF;
  float* cscale = w;             w += NSF;
  float* cshift = w;             w += NSF;
  int* deg     = (int*)w;        w += NE;       // also zeroed per call
  int* rowptr  = (int*)w;        w += NE + 1;
  int* cursor  = (int*)w;        w += NE;
  int* csr_src = (int*)w;        w += NLE;

  // 0) build line-graph CSR (by destination edge) -- removes 300M f32 atomics
  hipMemsetAsync(deg, 0, NE * sizeof(int), stream);
  lg_hist<<<CDIV(NLE, 256), 256, 0, stream>>>(lg1, deg);
  lg_scan<<<1, 1024, 0, stream>>>(deg, rowptr);
  hipMemcpyAsync(cursor, rowptr, NE * sizeof(int), hipMemcpyDeviceToDevice, stream);
  lg_fill<<<CDIV(NLE, 256), 256, 0, stream>>>(lg0, lg1, cursor, csr_src);

  // 1) initial edge features:  ea = ((x@Wu)[ei0] + (x@Wv)[ei1] + edge_attr@We)/3
  gemm(x, W_u, nullptr, XWu, NN, NF, NF, stream);
  gemm(x, W_v, nullptr, XWv, NN, NF, NF, stream);
  gemm(edge_attr, W_e, nullptr, ea, NE, NED, NF, stream);
  build_ea<<<CDIV(NE * NF, 256), 256, 0, stream>>>(XWu, XWv, ei0, ei1, ea);

  // 2) NITER rounds of line-graph message passing + GAT global pool
  const float* prev = ea;
  for (int t = 0; t < NITER; ++t) {
    float* out_t = outs + (size_t)t * NE * NF;
    gather_lg<<<CDIV(NE * NF, 256), 256, 0, stream>>>(prev, ea, out_t, rowptr, csr_src);

    edge_dot<<<CDIV(NE * 32, 256), 256, 0, stream>>>(out_t, gat_W, xw);
    fill_u32<<<CDIV(NE, 256), 256, 0, stream>>>(m_u, ORD_NEG_INF, NE);
    hipMemsetAsync(ssum, 0, NE * sizeof(float), stream);
    hipMemsetAsync(num, 0, NE * sizeof(float), stream);
    gat_max<<<CDIV(NLE + NE, 256), 256, 0, stream>>>(xw, att_src, att_dst, lg0, lg1, m_u);
    gat_sum<<<CDIV(NLE + NE, 256), 256, 0, stream>>>(xw, att_src, att_dst, lg0, lg1, m_u, ssum, num);

    fill_u32<<<CDIV(NB, 256), 256, 0, stream>>>(mb, ORD_NEG_INF, NB);
    hipMemsetAsync(sb, 0, NB * sizeof(float), stream);
    gat_xc_poolmax<<<CDIV(NE, 256), 256, 0, stream>>>(num, ssum, gat_bias, batch, xcb, mb);
    gat_poolw<<<CDIV(NE, 256), 256, 0, stream>>>(xcb, mb, batch, wexp, sb);

    hipMemsetAsync(gout_raw, 0, (size_t)NB * NF * sizeof(float), stream);
    gat_accum<<<CDIV(NE, 128), 256, 0, stream>>>(out_t, wexp, batch, gout_raw);
    gat_fin<<<CDIV(NB * NF, 256), 256, 0, stream>>>(gout_raw, sb);

    float* gout_t = gouts + (size_t)t * NB * NF;
    gemm(gout_raw, W_gout, b_gout, gout_t, NB, NF, NF, stream);
    tanh_k<<<CDIV(NB * NF, 256), 256, 0, stream>>>(gout_t, NB * NF);
    prev = out_t;
  }

  // 3) iteration attention mix + node scatter:  xn = x + segsum(sum_t out_t*sc, ei1)
  attn_sc<<<1, 256, 0, stream>>>(gouts, a_w, a_bias, scb);
  hipMemcpyAsync(xn, x, (size_t)NN * NF * sizeof(float), hipMemcpyDeviceToDevice, stream);
  final_edge_scatter<<<CDIV(NE * NF, 256), 256, 0, stream>>>(outs, scb, batch, ei1, xn);

  // 4) LinearBlock
  // lin1: fold bn1 into W1, X1 = xn @ W1f + b1f
  hipMemsetAsync(cs1, 0, NSF * sizeof(float), stream);
  hipMemsetAsync(cs2, 0, NSF * sizeof(float), stream);
  bn_stats<<<dim3(NF / 64, CDIV(NN, 128)), 256, 0, stream>>>(xn, NN, NF, cs1, cs2);
  bn_finalize<<<CDIV(NF, 256), 256, 0, stream>>>(cs1, cs2, bn1_g, bn1_b, NN, NF, cscale, cshift);
  fold_w1<<<CDIV(NSF, 256), 256, 0, stream>>>(cscale, cshift, W1, b1, W1f, b1f);
  gemm(xn, W1f, b1f, X1, NN, NF, NSF, stream);

  const int TOT = NN * NSF;
  // lin2: H = prelu(bn(X1)) @ Wm[0] + bm[0]
  hipMemsetAsync(cs1, 0, NSF * sizeof(float), stream);
  hipMemsetAsync(cs2, 0, NSF * sizeof(float), stream);
  bn_stats<<<dim3(NSF / 64, CDIV(NN, 128)), 256, 0, stream>>>(X1, NN, NSF, cs1, cs2);
  bn_finalize<<<CDIV(NSF, 256), 256, 0, stream>>>(cs1, cs2, bn_g, bn_b, NN, NSF, cscale, cshift);
  bn_prelu<<<CDIV(TOT, 256), 256, 0, stream>>>(X1, cscale, cshift, pr, 0, ACT, TOT, NSF);
  gemm(ACT, Wm, bm, Hb, NN, NSF, NSF, stream);

  // lin3: H = prelu(bn(H)) @ Wm[1] + bm[1];  X1 = (H + X1)/2
  hipMemsetAsync(cs1, 0, NSF * sizeof(float), stream);
  hipMemsetAsync(cs2, 0, NSF * sizeof(float), stream);
  bn_stats<<<dim3(NSF / 64, CDIV(NN, 128)), 256, 0, stream>>>(Hb, NN, NSF, cs1, cs2);
  bn_finalize<<<CDIV(NSF, 256), 256, 0, stream>>>(cs1, cs2, bn_g + NSF, bn_b + NSF, NN, NSF, cscale, cshift);
  bn_prelu<<<CDIV(TOT, 256), 256, 0, stream>>>(Hb, cscale, cshift, pr, 1, ACT, TOT, NSF);
  gemm(ACT, Wm + (size_t)NSF * NSF, bm + NSF, Hb, NN, NSF, NSF, stream);
  resid_avg<<<CDIV(TOT, 256), 256, 0, stream>>>(X1, Hb, TOT);

  // lin4: X1 = (prelu(bn(X1)) @ Wm[2] + bm[2] + X1)/2
  hipMemsetAsync(cs1, 0, NSF * sizeof(float), stream);
  hipMemsetAsync(cs2, 0, NSF * sizeof(float), stream);
  bn_stats<<<dim3(NSF / 64, CDIV(NN, 128)), 256, 0, stream>>>(X1, NN, NSF, cs1, cs2);
  bn_finalize<<<CDIV(NSF, 256), 256, 0, stream>>>(cs1, cs2, bn_g + 2 * NSF, bn_b + 2 * NSF, NN, NSF, cscale, cshift);
  bn_prelu<<<CDIV(TOT, 256), 256, 0, stream>>>(X1, cscale, cshift, pr, 2, ACT, TOT, NSF);
  gemm(ACT, Wm + (size_t)2 * NSF * NSF, bm + 2 * NSF, Hb, NN, NSF, NSF, stream);
  resid_avg<<<CDIV(TOT, 256), 256, 0, stream>>>(X1, Hb, TOT);

  // lin5: out = prelu(bn5(X1)) @ W5 + b5
  hipMemsetAsync(cs1, 0, NSF * sizeof(float), stream);
  hipMemsetAsync(cs2, 0, NSF * sizeof(float), stream);
  bn_stats<<<dim3(NSF / 64, CDIV(NN, 128)), 256, 0, stream>>>(X1, NN, NSF, cs1, cs2);
  bn_finalize<<<CDIV(NSF, 256), 256, 0, stream>>>(cs1, cs2, bn5_g, bn5_b, NN, NSF, cscale, cshift);
  bn_prelu<<<CDIV(TOT, 256), 256, 0, stream>>>(X1, cscale, cshift, pr5, 0, ACT, TOT, NSF);
  gemm(ACT, W5, b5, (float*)d_out, NN, NSF, NF, stream);
}